// kernel_generated_0_62904091017297
// MI455X (gfx1250) — compile-verified
//
#include <hip/hip_runtime.h>

// out[m,l,o,n] = C[m,(o-1)%56,n,l,0] + C[m,(o-2)%56,n,l,1]
// C[m,os,n,l,k] = sum_{j,i} xpad[m,j,os,n+i] * W[j,i,l,k]
// m=128, j=64, o=n=56, i=3, l=64, k=2.  fp32 -> V_WMMA_F32_16X16X4_F32.
//
// LDS layouts are K(j)-contiguous so every WMMA fragment is ONE ds_load_b64:
//   xs_t[s][col][j]   (A):  pair {xs[j][c], xs[j+1][c]} contiguous
//   Wl_t[k][i][l][j]  (B):  pair {W[j,..], W[j+1,..]}   contiguous
// j-stride 68 (== 4 mod 64): wave touches words 4c+j0+{0..3}, c=16 consecutive
// values -> 64 distinct banks -> conflict-free fragment loads.

typedef __attribute__((ext_vector_type(2))) float v2f;
typedef __attribute__((ext_vector_type(8))) float v8f;

#define JW 68                      // j-stride in floats
#define XCOLS 66                   // padded conv columns actually read (0..65)
#define WL_FLOATS (2*3*64*JW)      // Wl_t[k][i][l][JW]   = 26112
#define XS_FLOATS (2*XCOLS*JW)     // xs_t[s][col][JW]    = 8976
#define LDS_BYTES ((WL_FLOATS + XS_FLOATS)*4)

__global__ __launch_bounds__(256, 2)
void conv_shift_wmma_f32(const float* __restrict__ x,
                         const float* __restrict__ W,
                         float* __restrict__ out)
{
    extern __shared__ float smem[];
    float* Wl = smem;               // Wl_t[((k*3+i)*64+l)*JW + j] = W[j,i,l,k]
    float* xs = smem + WL_FLOATS;   // xs_t[(s*XCOLS+c)*JW + j] = xpad[m,j,os(s),c]

    const int tid = threadIdx.x;
    const int bid = blockIdx.x;
    const int m = bid / 56;
    const int o = bid % 56;
    const int os0 = (o + 55) % 56;  // slice feeding k=0
    const int os1 = (o + 54) % 56;  // slice feeding k=1

    // ---- stage W transposed:  W flat = ((j*3+i)*64+l)*2 + k ----
    for (int p = tid; p < 3*64*64; p += 256) {
        const int j = p / 192;
        const int i = (p / 64) % 3;
        const int l = p & 63;
        const float2 v = *(const float2*)(W + 2*p);      // {k=0, k=1}
        Wl[((0*3 + i)*64 + l)*JW + j] = v.x;
        Wl[((1*3 + i)*64 + l)*JW + j] = v.y;
    }

    // ---- stage the two x slices, zero-padded along n, transposed ----
    {
        const int row  = tid >> 1;          // 0..127 = s*64 + j
        const int half = tid & 1;
        const int s = row >> 6;
        const int j = row & 63;
        const int osrc = s ? os1 : os0;
        const float* src = x + (((m*64 + j)*56) + osrc)*56;   // 56 contiguous floats
        float* dst = xs + (s*XCOLS)*JW + j;                   // + c*JW per column
        if (half == 0) {
            dst[0] = 0.0f;                                    // c = 0
            #pragma unroll
            for (int q = 0; q < 14; ++q) {
                const float2 v = *(const float2*)(src + 2*q);
                dst[(1 + 2*q)*JW] = v.x;
                dst[(2 + 2*q)*JW] = v.y;
            }
        } else {
            #pragma unroll
            for (int c = 57; c < XCOLS; ++c) dst[c*JW] = 0.0f;
            #pragma unroll
            for (int q = 14; q < 28; ++q) {
                const float2 v = *(const float2*)(src + 2*q);
                dst[(1 + 2*q)*JW] = v.x;
                dst[(2 + 2*q)*JW] = v.y;
            }
        }
    }
    __syncthreads();

    // ---- WMMA GEMM: 8 waves, each owns 1 n-tile (16 rows) x 2 l-tiles ----
    const int w     = tid >> 5;
    const int lane  = tid & 31;
    const int laneM = lane & 15;    // A row M / B col N within tile
    const int laneH = lane >> 4;    // half-wave -> K += 2 (A/B), M += 8 (C/D)
    const int n0    = (w & 3) << 4;      // n-tile base: 0,16,32,48
    const int l0    = (w >> 2) << 5;     // l-tile base: 0 or 32 (two tiles each)

    v8f acc0 = {};
    v8f acc1 = {};

    for (int s = 0; s < 2; ++s) {
        for (int i = 0; i < 3; ++i) {
            // base addrs: all 16 unrolled steps use immediate DS offsets (+16B each)
            const float* pa = xs + (s*XCOLS + n0 + laneM + i)*JW + 2*laneH;
            const float* pb = Wl + ((s*3 + i)*64 + l0 + laneM)*JW + 2*laneH;
            #pragma unroll
            for (int j0 = 0; j0 < 64; j0 += 4) {
                const v2f a  = *(const v2f*)(pa + j0);            // ds_load_b64
                const v2f b0 = *(const v2f*)(pb + j0);            // ds_load_b64
                const v2f b1 = *(const v2f*)(pb + 16*JW + j0);    // ds_load_b64
                acc0 = __builtin_amdgcn_wmma_f32_16x16x4_f32(
                          false, a, false, b0, (short)0, acc0, false, false);
                acc1 = __builtin_amdgcn_wmma_f32_16x16x4_f32(
                          false, a, false, b1, (short)0, acc1, false, false);
            }
        }
    }

    // ---- store: lane's 8 acc values are n-contiguous -> two aligned float4 ----
    const int nBase = n0 + 8*laneH;
    if (nBase < 56) {                              // drop padded rows n=56..63
        const int l = l0 + laneM;
        float* p0 = out + ((m*64 + l)*56 + o)*56 + nBase;
        float* p1 = p0 + 16*56*56;                 // second l-tile (+16 channels)
        *(float4*)(p0)     = make_float4(acc0[0], acc0[1], acc0[2], acc0[3]);
        *(float4*)(p0 + 4) = make_float4(acc0[4], acc0[5], acc0[6], acc0[7]);
        *(float4*)(p1)     = make_float4(acc1[0], acc1[1], acc1[2], acc1[3]);
        *(float4*)(p1 + 4) = make_float4(acc1[4], acc1[5], acc1[6], acc1[7]);
    }
}

extern "C" void kernel_launch(void* const* d_in, const int* in_sizes, int n_in,
                              void* d_out, int out_size, void* d_ws, size_t ws_size,
                              hipStream_t stream)
{
    (void)in_sizes; (void)n_in; (void)out_size; (void)d_ws; (void)ws_size;
    const float* x = (const float*)d_in[0];   // (128,64,56,56)
    const float* W = (const float*)d_in[1];   // (64,3,64,2)
    float* out = (float*)d_out;               // (128,64,56,56)

    dim3 grid(128 * 56);
    dim3 block(256);
    conv_shift_wmma_f32<<<grid, block, LDS_BYTES, stream>>>(x, W, out);
}